// Net_link_2190433321525
// MI455X (gfx1250) — compile-verified
//
#include <hip/hip_runtime.h>

typedef __attribute__((ext_vector_type(2))) float v2f;
typedef __attribute__((ext_vector_type(8))) float v8f;

#define HID 128

// ---------------------------------------------------------------------------
// C[M x 128] = (RELU? max(A,0) : A)[M x 128] @ W[128 x 128]
// One wave computes a 16-row panel (16 x 128) via 8 WMMA f32 16x16x4 tiles.
// ---------------------------------------------------------------------------
template <bool RELU>
__global__ __launch_bounds__(256) void gemm128_wmma(
    const float* __restrict__ A, const float* __restrict__ W,
    float* __restrict__ C, int M)
{
    const int lane  = threadIdx.x & 31;
    const int wave  = threadIdx.x >> 5;
    const int m0    = (blockIdx.x * 8 + wave) * 16;
    const int nlane = lane & 15;
    const int halfK = (lane >> 4) * 2;   // 0 for lanes 0-15, 2 for lanes 16-31

    // Clamp A row for loads: WMMA requires EXEC all-1s, so no divergent exit.
    int arow = m0 + nlane;
    if (arow > M - 1) arow = M - 1;
    if (arow < 0) arow = 0;
    const float* Arow = A + (size_t)arow * HID;

    v8f acc[8];
#pragma unroll
    for (int t = 0; t < 8; ++t) acc[t] = (v8f)0.0f;

    for (int kk = 0; kk < HID; kk += 4) {
        v2f a;
        a.x = Arow[kk + halfK];
        a.y = Arow[kk + halfK + 1];
        if (RELU) { a.x = fmaxf(a.x, 0.0f); a.y = fmaxf(a.y, 0.0f); }
#pragma unroll
        for (int t = 0; t < 8; ++t) {
            const int col = t * 16 + nlane;
            v2f b;
            b.x = W[(size_t)(kk + halfK) * HID + col];
            b.y = W[(size_t)(kk + halfK + 1) * HID + col];
            acc[t] = __builtin_amdgcn_wmma_f32_16x16x4_f32(
                false, a, false, b, (short)0, acc[t], false, false);
        }
    }

    // C/D layout: VGPR r -> row m0 + r (lanes 0-15) or m0 + r + 8 (lanes 16-31)
    const int rbase = m0 + ((lane >> 4) ? 8 : 0);
#pragma unroll
    for (int r = 0; r < 8; ++r) {
        const int row = rbase + r;
        if (row < M) {
            float* Crow = C + (size_t)row * HID + nlane;
#pragma unroll
            for (int t = 0; t < 8; ++t) Crow[t * 16] = acc[t][r];
        }
    }
}

// ---------------------------------------------------------------------------
__global__ __launch_bounds__(256) void zero_kernel(float4* __restrict__ p, size_t n4)
{
    size_t i = (size_t)blockIdx.x * blockDim.x + threadIdx.x;
    if (i < n4) p[i] = make_float4(0.f, 0.f, 0.f, 0.f);
}

// ---------------------------------------------------------------------------
// out[dst] += w * xw[src] over E edges. One wave per edge: edge id in SGPR so
// index/weight loads are scalar; each lane gathers a float4 (b128) and issues
// 4 no-return global_atomic_add_f32 (L2-resident: 51.2MB << 192MB L2).
// ---------------------------------------------------------------------------
__global__ __launch_bounds__(256) void scatter_edges(
    const float* __restrict__ xw, const int* __restrict__ ei,
    const float* __restrict__ ew, float* __restrict__ out, int E)
{
    const int lane = threadIdx.x & 31;
    const int e = __builtin_amdgcn_readfirstlane(blockIdx.x * 8 + (threadIdx.x >> 5));
    if (e >= E) return;

    const int   s  = ei[e];       // edge_index[0][e]
    const int   d  = ei[E + e];   // edge_index[1][e]
    const float wt = ew[e];

    const float4 v = ((const float4*)(xw + (size_t)s * HID))[lane];
    float* dst = out + (size_t)d * HID + lane * 4;
    atomicAdd(dst + 0, v.x * wt);
    atomicAdd(dst + 1, v.y * wt);
    atomicAdd(dst + 2, v.z * wt);
    atomicAdd(dst + 3, v.w * wt);
}

// ---------------------------------------------------------------------------
// out[p,:] = concat(z[src[p]], z[dst[p]]) @ Wlin^T ; Wlin is [2, 256]
// ---------------------------------------------------------------------------
__global__ __launch_bounds__(256) void link_pred(
    const float* __restrict__ z, const int* __restrict__ pos,
    const float* __restrict__ Wlin, float* __restrict__ out, int P)
{
    __shared__ float wl[512];
    for (int i = threadIdx.x; i < 512; i += blockDim.x) wl[i] = Wlin[i];
    __syncthreads();

    const int p = blockIdx.x * blockDim.x + threadIdx.x;
    if (p >= P) return;

    const int s = pos[p];
    const int d = pos[P + p];
    const float4* zs = (const float4*)(z + (size_t)s * HID);
    const float4* zd = (const float4*)(z + (size_t)d * HID);

    float acc0 = 0.f, acc1 = 0.f;
#pragma unroll 4
    for (int k = 0; k < 32; ++k) {
        const float4 a = zs[k];
        const float* w0 = &wl[k * 4];           // row 0, cols 0..127
        const float* w1 = &wl[256 + k * 4];     // row 1, cols 0..127
        acc0 += a.x * w0[0] + a.y * w0[1] + a.z * w0[2] + a.w * w0[3];
        acc1 += a.x * w1[0] + a.y * w1[1] + a.z * w1[2] + a.w * w1[3];
    }
#pragma unroll 4
    for (int k = 0; k < 32; ++k) {
        const float4 b = zd[k];
        const float* w0 = &wl[128 + k * 4];         // row 0, cols 128..255
        const float* w1 = &wl[256 + 128 + k * 4];   // row 1, cols 128..255
        acc0 += b.x * w0[0] + b.y * w0[1] + b.z * w0[2] + b.w * w0[3];
        acc1 += b.x * w1[0] + b.y * w1[1] + b.z * w1[2] + b.w * w1[3];
    }
    out[(size_t)p * 2 + 0] = acc0;
    out[(size_t)p * 2 + 1] = acc1;
}

// ---------------------------------------------------------------------------
extern "C" void kernel_launch(void* const* d_in, const int* in_sizes, int n_in,
                              void* d_out, int out_size, void* d_ws, size_t ws_size,
                              hipStream_t stream)
{
    const float* x    = (const float*)d_in[0];
    const int*   ei1  = (const int*)  d_in[1];
    const int*   ei2  = (const int*)  d_in[2];
    const float* ew1  = (const float*)d_in[3];
    const float* ew2  = (const float*)d_in[4];
    const int*   pos  = (const int*)  d_in[5];
    const float* W1   = (const float*)d_in[6];
    const float* W2   = (const float*)d_in[7];
    const float* Wlin = (const float*)d_in[8];
    float*       out  = (float*)d_out;

    const int N  = in_sizes[0] / HID;
    const int E1 = in_sizes[1] / 2;
    const int E2 = in_sizes[2] / 2;
    const int P  = in_sizes[5] / 2;

    float* buf0 = (float*)d_ws;                 // xw1, then hw2
    float* buf1 = buf0 + (size_t)N * HID;       // h accum, then z accum

    const unsigned gemmBlocks = (unsigned)(((N + 15) / 16 + 7) / 8);
    const size_t   n4         = (size_t)N * HID / 4;
    const unsigned zeroBlocks = (unsigned)((n4 + 255) / 256);

    // 1) xw1 = x @ W1
    gemm128_wmma<false><<<gemmBlocks, 256, 0, stream>>>(x, W1, buf0, N);
    // 2) h = segment_sum(w1 * xw1[src1])
    zero_kernel<<<zeroBlocks, 256, 0, stream>>>((float4*)buf1, n4);
    scatter_edges<<<(unsigned)((E1 + 7) / 8), 256, 0, stream>>>(buf0, ei1, ew1, buf1, E1);
    // 3) hw2 = relu(h) @ W2   (ReLU fused into A-load)
    gemm128_wmma<true><<<gemmBlocks, 256, 0, stream>>>(buf1, W2, buf0, N);
    // 4) z = segment_sum(w2 * hw2[src2])
    zero_kernel<<<zeroBlocks, 256, 0, stream>>>((float4*)buf1, n4);
    scatter_edges<<<(unsigned)((E2 + 7) / 8), 256, 0, stream>>>(buf0, ei2, ew2, buf1, E2);
    // 5) out = concat(z[s], z[d]) @ Wlin^T
    link_pred<<<(unsigned)((P + 255) / 256), 256, 0, stream>>>(buf1, pos, Wlin, out, P);
}